// SapBERTembeddedSOM_13907104105187
// MI455X (gfx1250) — compile-verified
//
#include <hip/hip_runtime.h>
#include <stdint.h>

// ---------------------------------------------------------------------------
// SOM training step for MI455X (gfx1250, wave32).
// Both GEMMs via V_WMMA_F32_16X16X4_F32, with K-chunk double buffering in LDS
// fed by CDNA5 async-to-LDS loads (global_load_async_to_lds_b128, ASYNCcnt).
// ---------------------------------------------------------------------------

typedef float v2f __attribute__((ext_vector_type(2)));
typedef float v8f __attribute__((ext_vector_type(8)));

#define SOM_M  128
#define SOM_N  128
#define COMPS  (SOM_M * SOM_N)   // 16384
#define BATCH  4096
#define DIM    768
#define NITER  1000
#define ALPHA  0.3f
#define SIGMA  64.0f             // max(M,N)/2

#define KC       32              // K-chunk staged in LDS per pipeline stage
#define TS       36              // padded LDS row stride (floats): 16B aligned,
                                 // (36*l) % 64 distinct for l=0..15 -> no bank conflicts
#define TS4      68              // update-kernel X-tile stride (row-major B reads)

// monotone float -> uint mapping (total order matches float order)
__device__ __forceinline__ unsigned int forder(float f) {
  unsigned int u = __float_as_uint(f);
  return u ^ ((u >> 31) ? 0xFFFFFFFFu : 0x80000000u);
}

__device__ __forceinline__ unsigned long long shfl_xor_u64(unsigned long long v,
                                                           int mask, int width) {
  unsigned int lo = (unsigned int)v;
  unsigned int hi = (unsigned int)(v >> 32);
  lo = __shfl_xor(lo, mask, width);
  hi = __shfl_xor(hi, mask, width);
  return ((unsigned long long)hi << 32) | (unsigned long long)lo;
}

__device__ __forceinline__ unsigned long long u64min(unsigned long long a,
                                                     unsigned long long b) {
  return a < b ? a : b;
}

// async copy of 16 bytes: global -> LDS, tracked by ASYNCcnt (CDNA5)
__device__ __forceinline__ void async_g2l_b128(unsigned lds_off,
                                               unsigned long long gaddr) {
  asm volatile("global_load_async_to_lds_b128 %0, %1, off"
               :: "v"(lds_off), "v"(gaddr) : "memory");
}

// ---------------------------------------------------------------------------
// Kernel 1: squared row norms of X (4096 rows) and W (16384 rows).
// ---------------------------------------------------------------------------
__global__ void __launch_bounds__(256)
som_norms_kernel(const float* __restrict__ X, const float* __restrict__ W,
                 float* __restrict__ xnorm, float* __restrict__ wnorm) {
  int wave = (blockIdx.x * blockDim.x + threadIdx.x) >> 5;
  int lane = threadIdx.x & 31;
  if (wave >= BATCH + COMPS) return;
  const float* row = (wave < BATCH) ? (X + (size_t)wave * DIM)
                                    : (W + (size_t)(wave - BATCH) * DIM);
  float s = 0.0f;
  for (int i = lane * 4; i < DIM; i += 32 * 4) {
    float4 v = *(const float4*)(row + i);
    s += v.x * v.x + v.y * v.y + v.z * v.z + v.w * v.w;
  }
  for (int m = 16; m >= 1; m >>= 1) s += __shfl_xor(s, m, 32);
  if (lane == 0) {
    if (wave < BATCH) xnorm[wave] = s;
    else              wnorm[wave - BATCH] = s;
  }
}

// ---------------------------------------------------------------------------
// Kernel 2: reset BMU argmin keys.
// ---------------------------------------------------------------------------
__global__ void som_bmu_init_kernel(unsigned long long* __restrict__ bmu) {
  int i = blockIdx.x * blockDim.x + threadIdx.x;
  if (i < BATCH) bmu[i] = 0xFFFFFFFFFFFFFFFFull;
}

// ---------------------------------------------------------------------------
// Kernel 3: BMU search.  S = X * W^T via V_WMMA_F32_16X16X4_F32.
// Block tile 32(b) x 256(c); 8 waves as 2x4, each wave 16x64.
// K pipelined in KC=32 chunks: async-to-LDS double buffer, fragments from LDS.
// Fused epilogue: d2 -> (orderable-dist,col) u64 key -> half-wave shfl min
// -> atomicMin per batch row.
// ---------------------------------------------------------------------------
__global__ void __launch_bounds__(256)
som_bmu_kernel(const float* __restrict__ X, const float* __restrict__ W,
               const float* __restrict__ xnorm, const float* __restrict__ wnorm,
               unsigned long long* __restrict__ bmu) {
  __shared__ float Xs[2][32][TS];    //  9.2 KB
  __shared__ float Ws[2][256][TS];   // 73.7 KB

  const int tid  = threadIdx.x;
  const int lane = tid & 31;
  const int wid  = tid >> 5;          // 0..7
  const int half = lane >> 4;
  const int lx   = lane & 15;
  const int bBase = blockIdx.y * 32;
  const int cBase = blockIdx.x * 256;
  const int bOff  = (wid & 1) * 16;   // wave tile inside block
  const int cOff  = (wid >> 1) * 64;

  // cooperative staging coords: one b128 per thread per 32x32 sub-tile
  const int srow = tid >> 3;          // 0..31
  const int scol = (tid & 7) * 4;     // 0..28

  const unsigned long long xg =
      (unsigned long long)(uintptr_t)(X + (size_t)(bBase + srow) * DIM + scol);
  const unsigned long long wg =
      (unsigned long long)(uintptr_t)(W + (size_t)(cBase + srow) * DIM + scol);
  const unsigned xl[2] = {(unsigned)(uintptr_t)&Xs[0][srow][scol],
                          (unsigned)(uintptr_t)&Xs[1][srow][scol]};
  const unsigned wl[2] = {(unsigned)(uintptr_t)&Ws[0][srow][scol],
                          (unsigned)(uintptr_t)&Ws[1][srow][scol]};

  auto stage = [&](int chunk, int buf) {
    const unsigned long long koff = (unsigned long long)chunk * (KC * 4);
    async_g2l_b128(xl[buf], xg + koff);           // X: 32 rows
#pragma unroll
    for (int r = 0; r < 8; ++r) {                 // W: 256 rows (8 x 32)
      async_g2l_b128(wl[buf] + r * (32 * TS * 4),
                     wg + koff + (unsigned long long)r * (32 * DIM * 4));
    }
  };

  stage(0, 0);

  v8f acc[4] = {};
  const int NCHUNK = DIM / KC;        // 24
  for (int ch = 0; ch < NCHUNK; ++ch) {
    const int buf = ch & 1;
    if (ch + 1 < NCHUNK) {
      stage(ch + 1, buf ^ 1);
      asm volatile("s_wait_asynccnt 0x9" ::: "memory");  // current chunk done
    } else {
      asm volatile("s_wait_asynccnt 0x0" ::: "memory");
    }
    __syncthreads();
#pragma unroll
    for (int kk = 0; kk < KC; kk += 4) {
      v2f a = *(const v2f*)&Xs[buf][bOff + lx][kk + half * 2];
#pragma unroll
      for (int t = 0; t < 4; ++t) {
        v2f b = *(const v2f*)&Ws[buf][cOff + t * 16 + lx][kk + half * 2];
        acc[t] = __builtin_amdgcn_wmma_f32_16x16x4_f32(
            false, a, false, b, (short)0, acc[t], false, false);
      }
    }
    __syncthreads();                  // protect buf before it is re-staged
  }

  float wn[4];
#pragma unroll
  for (int t = 0; t < 4; ++t) wn[t] = wnorm[cBase + cOff + t * 16 + lx];

#pragma unroll
  for (int g = 0; g < 8; ++g) {
    const int brow_idx = bBase + bOff + g + half * 8;   // C/D layout row
    const float xn = xnorm[brow_idx];
    unsigned long long best = 0xFFFFFFFFFFFFFFFFull;
#pragma unroll
    for (int t = 0; t < 4; ++t) {
      int c = cBase + cOff + t * 16 + lx;
      float d2 = xn + wn[t] - 2.0f * acc[t][g];
      unsigned long long key =
          ((unsigned long long)forder(d2) << 32) | (unsigned int)c;
      best = u64min(best, key);
    }
#pragma unroll
    for (int m = 8; m >= 1; m >>= 1)
      best = u64min(best, shfl_xor_u64(best, m, 16));
    if (lx == 0) atomicMin(&bmu[brow_idx], best);
  }
}

// ---------------------------------------------------------------------------
// Kernel 4: neighbourhood update.
//   out[c,:] = W[c,:] * (1 - L[c]) + sum_b lr[b,c] * X[b,:]
// A = lr generated in-registers (v_exp_f32), B = X staged through an
// async-to-LDS double buffer shared by all 8 waves (block tile 128c x 64n).
// BMU locations decoded once into LDS; column sums L[c] fall out of A.
// ---------------------------------------------------------------------------
__global__ void __launch_bounds__(256)
som_update_kernel(const float* __restrict__ X, const float* __restrict__ W,
                  const unsigned long long* __restrict__ bmu,
                  const int* __restrict__ itp, float* __restrict__ out) {
  __shared__ float2 sb[BATCH];        // 32 KB decoded BMU (i,j)
  __shared__ float  Bs[2][KC][TS4];   // 17.4 KB X tile double buffer
  __shared__ float  sL[8][16];

  const int tid  = threadIdx.x;
  const int lane = tid & 31;
  const int wid  = tid >> 5;
  const int half = lane >> 4;
  const int lx   = lane & 15;

  const int cg    = blockIdx.x / 12;          // 128 c-groups
  const int nt    = blockIdx.x % 12;          // 12 n-tiles
  const int cBase = cg * 128 + wid * 16;      // per-wave 16 c-rows
  const int nBase = nt * 64;                  // shared 64 n-cols

  for (int i = tid; i < BATCH; i += blockDim.x) {
    unsigned int idx = (unsigned int)(bmu[i] & 0xFFFFFFFFu);
    sb[i] = make_float2((float)(idx >> 7), (float)(idx & (SOM_N - 1)));
  }

  // staging coords: 32 rows x 64 cols = 2 b128 per thread
  const int scol = (tid & 15) * 4;            // 0..60
  const int srow = tid >> 4;                  // 0..15 (+16 on 2nd issue)
  const unsigned long long bg =
      (unsigned long long)(uintptr_t)(X + (size_t)srow * DIM + nBase + scol);
  const unsigned bl[2] = {(unsigned)(uintptr_t)&Bs[0][srow][scol],
                          (unsigned)(uintptr_t)&Bs[1][srow][scol]};

  auto stage = [&](int chunk, int buf) {
    const unsigned long long koff = (unsigned long long)chunk * (KC * DIM * 4);
    async_g2l_b128(bl[buf], bg + koff);
    async_g2l_b128(bl[buf] + 16 * TS4 * 4,
                   bg + koff + (unsigned long long)16 * DIM * 4);
  };

  stage(0, 0);

  const float decay    = 1.0f - (float)itp[0] / (float)NITER;
  const float alpha_op = ALPHA * decay;
  const float sigma_op = SIGMA * decay;
  const float inv_ss   = 1.0f / (sigma_op * sigma_op);

  const int   myc = cBase + lx;               // A row (both halves)
  const float ci  = (float)(myc >> 7);
  const float cj  = (float)(myc & (SOM_N - 1));

  v8f   acc[4] = {};
  float Lsum   = 0.0f;

  const int NCHUNK = BATCH / KC;              // 128
  for (int ch = 0; ch < NCHUNK; ++ch) {
    const int buf = ch & 1;
    if (ch + 1 < NCHUNK) {
      stage(ch + 1, buf ^ 1);
      asm volatile("s_wait_asynccnt 0x2" ::: "memory");
    } else {
      asm volatile("s_wait_asynccnt 0x0" ::: "memory");
    }
    __syncthreads();
#pragma unroll
    for (int kk = 0; kk < KC; kk += 4) {
      const int b0 = ch * KC + kk + half * 2;  // this lane's two batch indices
      float2 p0 = sb[b0];
      float2 p1 = sb[b0 + 1];
      float di0 = ci - p0.x, dj0 = cj - p0.y;
      float di1 = ci - p1.x, dj1 = cj - p1.y;
      v2f a;
      a.x = alpha_op * __expf(-(di0 * di0 + dj0 * dj0) * inv_ss);
      a.y = alpha_op * __expf(-(di1 * di1 + dj1 * dj1) * inv_ss);
      Lsum += a.x + a.y;
      const int r0 = kk + half * 2;
#pragma unroll
      for (int t = 0; t < 4; ++t) {
        v2f b;
        b.x = Bs[buf][r0][t * 16 + lx];
        b.y = Bs[buf][r0 + 1][t * 16 + lx];
        acc[t] = __builtin_amdgcn_wmma_f32_16x16x4_f32(
            false, a, false, b, (short)0, acc[t], false, false);
      }
    }
    __syncthreads();
  }

  // L[c] = partial(lanes 0-15) + partial(lanes 16-31) for the same c
  Lsum += __shfl_xor(Lsum, 16, 32);
  if (half == 0) sL[wid][lx] = Lsum;
  __syncthreads();

#pragma unroll
  for (int t = 0; t < 4; ++t) {
    const int n = nBase + t * 16 + lx;
#pragma unroll
    for (int g = 0; g < 8; ++g) {
      const int   c = cBase + g + half * 8;   // C/D layout row
      const float L = sL[wid][g + half * 8];
      const size_t o = (size_t)c * DIM + n;
      out[o] = W[o] * (1.0f - L) + acc[t][g];
    }
  }
}

// ---------------------------------------------------------------------------
extern "C" void kernel_launch(void* const* d_in, const int* in_sizes, int n_in,
                              void* d_out, int out_size, void* d_ws, size_t ws_size,
                              hipStream_t stream) {
  const float* X   = (const float*)d_in[0];   // [4096, 768]
  const float* W   = (const float*)d_in[1];   // [16384, 768]
  const int*   itp = (const int*)d_in[2];     // scalar
  float*       out = (float*)d_out;           // [16384, 768]

  char* ws = (char*)d_ws;
  float* xnorm = (float*)ws;                                   // 16 KB
  float* wnorm = (float*)(ws + BATCH * sizeof(float));         // 64 KB
  unsigned long long* bmu =
      (unsigned long long*)(ws + (BATCH + COMPS) * sizeof(float)); // 32 KB

  som_norms_kernel<<<(BATCH + COMPS) / 8, 256, 0, stream>>>(X, W, xnorm, wnorm);
  som_bmu_init_kernel<<<(BATCH + 255) / 256, 256, 0, stream>>>(bmu);
  som_bmu_kernel<<<dim3(COMPS / 256, BATCH / 32), 256, 0, stream>>>(
      X, W, xnorm, wnorm, bmu);
  som_update_kernel<<<128 * 12, 256, 0, stream>>>(X, W, bmu, itp, out);
}